// _LiftSplat_63359357551094
// MI455X (gfx1250) — compile-verified
//
#include <hip/hip_runtime.h>
#include <math.h>

// ---------------- problem constants (match reference setup) ----------------
#define B_   2
#define TT_  2
#define C6   6
#define CR   64
#define HP   56
#define WP   100
#define H_   448
#define W_   800
#define NCAM (B_ * TT_ * C6)     // 24
#define NBT  (B_ * TT_)          // 4
#define P_   (HP * WP)           // 5600
#define HB   200
#define WB   200
#define NBINS (HB * WB)          // 40000
#define X_MINF (-50.0f)
#define X_MAXF (50.0f)
#define Y_MINF (-50.0f)
#define Y_MAXF (50.0f)
#define RX_INV (2.0f)            // 1 / ((X_MAX-X_MIN)/WB) = 1/0.5
#define RY_INV (2.0f)

// workspace layout (float elements)
#define ACC_OFF   0                                  // NBT*NBINS*CR = 10,240,000
#define ACC_ELEMS ((size_t)NBT * NBINS * CR)
#define CNT_OFF   (ACC_OFF + ACC_ELEMS)              // NCAM*NBINS = 960,000
#define CNT_ELEMS ((size_t)NCAM * NBINS)
#define CSUM_OFF  (CNT_OFF + CNT_ELEMS)              // NBT*NBINS = 160,000
#define CSUM_ELEMS ((size_t)NBT * NBINS)
#define CAMP_OFF  (CSUM_OFF + CSUM_ELEMS)            // NCAM*16 = 384
#define CAMP_ELEMS ((size_t)NCAM * 16)
#define PBIN_OFF  (CAMP_OFF + CAMP_ELEMS)            // NCAM*P_ ints = 134,400

typedef float v2f __attribute__((ext_vector_type(2)));
typedef float v8f __attribute__((ext_vector_type(8)));

// ---------------- kernel 0: zero acc + cnt ----------------
__global__ __launch_bounds__(256) void ls_zero(float* ws_f, int total) {
    int i = blockIdx.x * blockDim.x + threadIdx.x;
    if (i < total) ws_f[i] = 0.0f;
}

// ---------------- kernel 1: per-camera params (fx,fy,cx,cy, Tinv 3x4) ----------------
__global__ void ls_prep(const float* __restrict__ Ks, const float* __restrict__ T,
                        float* __restrict__ camp) {
    int n = threadIdx.x;
    if (n >= NCAM) return;
    const float* Kn = Ks + n * 9;
    const float* Tn = T + n * 16;
    const float Sx = (float)WP / (float)W_;   // 0.125
    const float Sy = (float)HP / (float)H_;   // 0.125
    float* cp = camp + n * 16;
    cp[0] = Kn[0] * Sx;   // fx
    cp[1] = Kn[4] * Sy;   // fy
    cp[2] = Kn[2] * Sx;   // cx
    cp[3] = Kn[5] * Sy;   // cy
    // affine inverse of 4x4 (bottom row [0 0 0 1]): [R^-1 | -R^-1 t]
    float r00=Tn[0],  r01=Tn[1],  r02=Tn[2],  tx=Tn[3];
    float r10=Tn[4],  r11=Tn[5],  r12=Tn[6],  ty=Tn[7];
    float r20=Tn[8],  r21=Tn[9],  r22=Tn[10], tz=Tn[11];
    float c00 =  (r11*r22 - r12*r21);
    float c10 = -(r10*r22 - r12*r20);
    float c20 =  (r10*r21 - r11*r20);
    float det = r00*c00 + r01*c10 + r02*c20;
    float id  = 1.0f / det;
    float i00 = c00*id, i01 = -(r01*r22 - r02*r21)*id, i02 =  (r01*r12 - r02*r11)*id;
    float i10 = c10*id, i11 =  (r00*r22 - r02*r20)*id, i12 = -(r00*r12 - r02*r10)*id;
    float i20 = c20*id, i21 = -(r00*r21 - r01*r20)*id, i22 =  (r00*r11 - r01*r10)*id;
    cp[4]=i00;  cp[5]=i01;  cp[6]=i02;  cp[7]  = -(i00*tx + i01*ty + i02*tz);
    cp[8]=i10;  cp[9]=i11;  cp[10]=i12; cp[11] = -(i10*tx + i11*ty + i12*tz);
    cp[12]=i20; cp[13]=i21; cp[14]=i22; cp[15] = -(i20*tx + i21*ty + i22*tz);
}

// ---------------- kernel 2: WMMA projection + binning + per-camera count ----------------
// Each wave32 transforms 16 points with one V_WMMA_F32_16X16X4_F32:
//   A (16x4, rows 0..2 = Tinv[:3,:4])  x  B (4x16, col n = [x_c, y_c, d, 1] of point n)
// D row0 = x_e, row1 = y_e for 16 points (lanes 0..15).
__global__ __launch_bounds__(256) void ls_proj(const float* __restrict__ depths,
                                               const float* __restrict__ camp,
                                               float* __restrict__ cnt,
                                               int* __restrict__ pbin) {
    const int NP = NCAM * P_;
    int tid  = blockIdx.x * blockDim.x + threadIdx.x;
    int wave = tid >> 5;
    int lane = tid & 31;
    int nn   = lane & 15;     // row of A / column of B / point-in-group
    int hi   = lane >> 4;     // 0 -> K0,K1 ; 1 -> K2,K3
    int qbase = wave << 4;
    if (qbase >= NP) return;  // wave-uniform: EXEC stays all-ones for WMMA

    int q  = qbase + nn;      // global point id; P_ % 16 == 0 -> one camera per wave
    int n  = q / P_;
    int p  = q - n * P_;
    int hp = p / WP;
    int wp = p - hp * WP;

    // depth gather: src_y = hp*8, src_x = wp*8 (floor(hp*448/56), floor(wp*800/100))
    float d = depths[(size_t)n * (H_ * W_) + (size_t)(hp * 8) * W_ + (wp * 8)];

    const float* cp = camp + n * 16;
    float fx = cp[0], fy = cp[1], cx = cp[2], cy = cp[3];

    // A operand: lane<16 holds Tinv[m][0],Tinv[m][1]; lane>=16 holds Tinv[m][2],Tinv[m][3]
    v2f a;
    if (nn < 3) {
        a.x = cp[4 + nn * 4 + (hi ? 2 : 0)];
        a.y = cp[4 + nn * 4 + (hi ? 3 : 1)];
    } else {
        a.x = 0.0f; a.y = 0.0f;
    }

    // B operand: lane<16 holds (x_c, y_c); lane>=16 holds (d, 1)
    float xc = ((float)wp - cx) / fx * d;
    float yc = ((float)hp - cy) / fy * d;
    v2f b;
    if (hi == 0) { b.x = xc; b.y = yc; }
    else         { b.x = d;  b.y = 1.0f; }

    v8f c = {};
#if __has_builtin(__builtin_amdgcn_wmma_f32_16x16x4_f32)
    c = __builtin_amdgcn_wmma_f32_16x16x4_f32(false, a, false, b, (short)0, c,
                                              false, false);
    float xe = c[0];   // D row 0, col nn (lanes 0..15)
    float ye = c[1];   // D row 1, col nn (lanes 0..15)
#else
#warning "wmma_f32_16x16x4_f32 builtin missing - scalar fallback in use"
    float xe = cp[4]  * xc + cp[5]  * yc + cp[6]  * d + cp[7];
    float ye = cp[8]  * xc + cp[9]  * yc + cp[10] * d + cp[11];
#endif

    if (hi == 0) {
        bool valid = (d > 0.0f) & (xe >= X_MINF) & (xe < X_MAXF) &
                     (ye >= Y_MINF) & (ye < Y_MAXF);
        int lin = -1;
        if (valid) {
            int ix = (int)floorf((xe - X_MINF) * RX_INV);
            int iy = (int)floorf((ye - Y_MINF) * RY_INV);
            ix = min(max(ix, 0), WB - 1);
            iy = min(max(iy, 0), HB - 1);
            lin = iy * WB + ix;
            atomicAdd(&cnt[n * NBINS + lin], 1.0f);
        }
        pbin[q] = lin;
    }
}

// ---------------- kernel 3: channel scatter (feats read fully coalesced) ----------------
__global__ __launch_bounds__(256) void ls_scatter(const float* __restrict__ feats,
                                                  const int* __restrict__ pbin,
                                                  float* __restrict__ acc) {
    int idx = blockIdx.x * blockDim.x + threadIdx.x;   // == flat feats index (n, cr, p)
    const int CRP = CR * P_;
    int n  = idx / CRP;
    int r  = idx - n * CRP;
    int cr = r / P_;
    int p  = r - cr * P_;
    int bin = pbin[n * P_ + p];
    if (bin >= 0) {
        float f = feats[idx];
        int bt = n / C6;
        atomicAdd(&acc[((size_t)bt * NBINS + bin) * CR + cr], f);
    }
}

// ---------------- kernel 4: cnt_total(bt,bin) = sum_c max(cnt, 1) ----------------
__global__ __launch_bounds__(256) void ls_cntsum(const float* __restrict__ cnt,
                                                 float* __restrict__ csum) {
    int idx = blockIdx.x * blockDim.x + threadIdx.x;
    if (idx >= NBT * NBINS) return;
    int bt  = idx / NBINS;
    int bin = idx - bt * NBINS;
    float s = 0.0f;
    #pragma unroll
    for (int c = 0; c < C6; ++c)
        s += fmaxf(cnt[(bt * C6 + c) * NBINS + bin], 1.0f);
    csum[idx] = s;
}

// ---------------- kernel 5: divide + (bin,cr)->(cr,bin) transpose via LDS tile ----------
__global__ __launch_bounds__(256) void ls_final(const float* __restrict__ acc,
                                                const float* __restrict__ csum,
                                                float* __restrict__ out) {
    __shared__ float tile[64][65];                    // pad: no bank conflicts
    const int TILES = NBINS / 64;                     // 625
    int bt   = blockIdx.x / TILES;
    int bin0 = (blockIdx.x - bt * TILES) * 64;
    int tid  = threadIdx.x;
    #pragma unroll
    for (int i = 0; i < 16; ++i) {                    // read coalesced over cr
        int lin = tid + i * 256;
        int bl  = lin >> 6;
        int cr  = lin & 63;
        float v = acc[((size_t)bt * NBINS + bin0 + bl) * CR + cr];
        tile[bl][cr] = v / csum[bt * NBINS + bin0 + bl];
    }
    __syncthreads();
    #pragma unroll
    for (int i = 0; i < 16; ++i) {                    // write coalesced over bin
        int lin = tid + i * 256;
        int crl = lin >> 6;
        int bl  = lin & 63;
        out[((size_t)(bt * CR + crl)) * NBINS + bin0 + bl] = tile[bl][crl];
    }
}

// ---------------- launcher ----------------
extern "C" void kernel_launch(void* const* d_in, const int* in_sizes, int n_in,
                              void* d_out, int out_size, void* d_ws, size_t ws_size,
                              hipStream_t stream) {
    const float* feats  = (const float*)d_in[0];
    const float* depths = (const float*)d_in[1];
    const float* Ks     = (const float*)d_in[2];
    const float* Tcam   = (const float*)d_in[3];
    (void)in_sizes; (void)n_in; (void)out_size; (void)ws_size;

    float* ws_f = (float*)d_ws;
    float* acc  = ws_f + ACC_OFF;
    float* cnt  = ws_f + CNT_OFF;
    float* csum = ws_f + CSUM_OFF;
    float* camp = ws_f + CAMP_OFF;
    int*   pbin = (int*)(ws_f + PBIN_OFF);
    float* out  = (float*)d_out;

    // 0) zero acc + cnt (contiguous in ws)
    int zero_total = (int)(ACC_ELEMS + CNT_ELEMS);    // 11,200,000
    ls_zero<<<(zero_total + 255) / 256, 256, 0, stream>>>(ws_f, zero_total);

    // 1) per-camera params
    ls_prep<<<1, 32, 0, stream>>>(Ks, Tcam, camp);

    // 2) WMMA projection: 24*5600/16 = 8400 waves = 1050 blocks of 8 waves
    ls_proj<<<(NCAM * P_ / 16) / 8, 256, 0, stream>>>(depths, camp, cnt, pbin);

    // 3) channel scatter: 24*64*5600 = 8,601,600 threads (exact multiple of 256)
    ls_scatter<<<(NCAM * CR * P_) / 256, 256, 0, stream>>>(feats, pbin, acc);

    // 4) per-bin count totals
    ls_cntsum<<<(NBT * NBINS + 255) / 256, 256, 0, stream>>>(cnt, csum);

    // 5) divide + transpose to (B,Tt,Cr,HB,WB)
    ls_final<<<NBT * (NBINS / 64), 256, 0, stream>>>(acc, csum, out);
}